// SimDecoder_85624468013473
// MI455X (gfx1250) — compile-verified
//
#include <hip/hip_runtime.h>
#include <math.h>

typedef float v2f __attribute__((ext_vector_type(2)));
typedef float v8f __attribute__((ext_vector_type(8)));

#define NN 30
#define FF 4
#define HH 64
#define EE 870           // N*(N-1)
#define BCC 2048         // (B*S)*C = 32*64
#define ROWS (BCC*NN)    // 61440
#define OUT_HALF (8*4*127*30*4)  // 487680

// D = A(16x4,f32) * B(4x16,f32) + C(16x16,f32), full-precision WMMA
__device__ __forceinline__ v8f wmma4(v2f a, v2f b, v8f c) {
  return __builtin_amdgcn_wmma_f32_16x16x4_f32(false, a, false, b, (short)0, c, false, false);
}

// ---------------------------------------------------------------------------
// Kernel 1: prev[bc][n][f] = data[bs][t=2c][n][f]   (bc = bs*64 + c)
// ---------------------------------------------------------------------------
__global__ void prep_prev(const float* __restrict__ data, float* __restrict__ prev) {
  int idx = blockIdx.x * blockDim.x + threadIdx.x;
  if (idx >= BCC * NN * FF) return;
  int f  = idx & 3;
  int n  = (idx >> 2) % NN;
  int bc = idx / (NN * FF);
  int c  = bc & 63;
  int bs = bc >> 6;
  prev[idx] = data[((bs * 128 + 2 * c) * NN + n) * FF + f];
}

// ---------------------------------------------------------------------------
// Kernel 2: per-(bc) fused edge MLP + graph mask + scatter-add into agg.
//   h1(i,j) = relu(Pr[i] + Ps[j])  with Pr = prev@W1[0:4]+b1, Ps = prev@W1[4:8]
//   msg     = h1 @ W2 + b2,  msg *= graph[bs][e],  agg[recv] += msg
// Edge GEMM2 (870x64x64) done with V_WMMA_F32_16X16X4_F32.
// Fully branchless inner loops: Pr/Ps/aggS padded to 31 rows, gS padded to
// 880 with zero tail, so out-of-range edge lanes are masked by g=0 and land
// in the discarded pad row.  EXEC stays all-1s around every WMMA.
// ---------------------------------------------------------------------------
__global__ void __launch_bounds__(256)
edge_agg(const float* __restrict__ prev,
         const float* __restrict__ graph,
         const float* __restrict__ msgW1, const float* __restrict__ msgB1,
         const float* __restrict__ msgW2, const float* __restrict__ msgB2,
         float* __restrict__ agg)
{
  __shared__ float Pr[(NN + 1) * HH];   // +1 pad row
  __shared__ float Ps[(NN + 1) * HH];   // +1 pad row
  __shared__ float W2s[HH * HH];
  __shared__ float aggS[(NN + 1) * HH]; // +1 pad row (absorbs e>=870)
  __shared__ float gS[880];             // zero-padded tail masks e>=870
  __shared__ float b2s[HH];
  __shared__ float pv[NN * FF];

  const int tid = threadIdx.x;
  const int bc  = blockIdx.x;
  const int bs  = bc >> 6;

  for (int i = tid; i < HH * HH; i += 256) W2s[i] = msgW2[i];
  for (int i = tid; i < 880; i += 256)     gS[i]  = (i < EE) ? graph[bs * EE + i] : 0.f;
  if (tid < HH)                            b2s[tid] = msgB2[tid];
  for (int i = tid; i < NN * FF; i += 256) pv[i]  = prev[bc * NN * FF + i];
  for (int i = tid; i < (NN + 1) * HH; i += 256) aggS[i] = 0.f;
  if (tid < HH) { Pr[NN * HH + tid] = 0.f; Ps[NN * HH + tid] = 0.f; }  // pad row
  __syncthreads();

  // Pr/Ps precompute (rank-structure of the K=8 first layer)
  for (int i = tid; i < NN * HH; i += 256) {
    int n = i >> 6, h = i & 63;
    float pr = msgB1[h], ps = 0.f;
#pragma unroll
    for (int f = 0; f < FF; ++f) {
      float x = pv[n * FF + f];
      pr += x * msgW1[f * HH + h];
      ps += x * msgW1[(FF + f) * HH + h];
    }
    Pr[i] = pr;
    Ps[i] = ps;
  }
  __syncthreads();

  const int lane = tid & 31;
  const int wave = tid >> 5;
  const int half = lane >> 4;   // K-half for A/B fragments
  const int l15  = lane & 15;

  // 55 tiles of 16 edges (tile 54 partial; pad rows absorb edges 870..879)
  for (int t = wave; t < 55; t += 8) {
    const int eA = t * 16 + l15;            // <= 879, fits padded Pr/Ps
    const int iA = eA / 29;                 // <= 30 (pad row)
    const int jj = eA - iA * 29;
    const int jA = jj + (jj >= iA ? 1 : 0); // <= 29

    v8f acc[4] = {v8f{}, v8f{}, v8f{}, v8f{}};

#pragma unroll
    for (int kk = 0; kk < 16; ++kk) {
      const int k0 = kk * 4 + half * 2;
      v2f a;
      a.x = fmaxf(Pr[iA * HH + k0]     + Ps[jA * HH + k0],     0.f);
      a.y = fmaxf(Pr[iA * HH + k0 + 1] + Ps[jA * HH + k0 + 1], 0.f);
#pragma unroll
      for (int nt = 0; nt < 4; ++nt) {
        v2f b;
        b.x = W2s[k0 * HH       + nt * 16 + l15];
        b.y = W2s[(k0 + 1) * HH + nt * 16 + l15];
        acc[nt] = wmma4(a, b, acc[nt]);
      }
    }

    // bias + graph mask + segmented scatter-add (LDS float atomics),
    // branchless: g=0 kills invalid edges, pad row absorbs their adds.
#pragma unroll
    for (int r = 0; r < 8; ++r) {
      const int e  = t * 16 + r + half * 8;  // C-matrix VGPR r -> M = r (+8)
      const int ii = e / 29;                 // RECV node (edges sorted by recv)
      const float g = gS[e];
#pragma unroll
      for (int nt = 0; nt < 4; ++nt) {
        const int col = nt * 16 + l15;
        atomicAdd(&aggS[ii * HH + col], (acc[nt][r] + b2s[col]) * g);
      }
    }
  }
  __syncthreads();
  for (int i = tid; i < NN * HH; i += 256) agg[bc * NN * HH + i] = aggS[i];
}

// ---------------------------------------------------------------------------
// Kernel 3: node MLPs. 32 flat rows (bc*30+n) per block.
//   mean: agg->64 relu ->64 relu ->4            (prevNext + out)
//   var : agg->64 relu ->64 relu ->4 softplus   (out)
// Two WMMA GEMM layers per path (32x64x64 each), ping-pong weights in LDS.
// ---------------------------------------------------------------------------
__global__ void __launch_bounds__(256)
node_mlp(const float* __restrict__ agg,
         const float* __restrict__ mW1, const float* __restrict__ mB1,
         const float* __restrict__ mW2, const float* __restrict__ mB2,
         const float* __restrict__ mW3, const float* __restrict__ mB3,
         const float* __restrict__ vW1, const float* __restrict__ vB1,
         const float* __restrict__ vW2, const float* __restrict__ vB2,
         const float* __restrict__ vW3, const float* __restrict__ vB3,
         float* __restrict__ out, float* __restrict__ prevNext, int step)
{
  __shared__ float aggT[32 * HH];
  __shared__ float hA[32 * HH];
  __shared__ float hB[32 * HH];
  __shared__ float Wa[HH * HH];
  __shared__ float Wb[HH * HH];
  __shared__ float W3s[HH * FF];
  __shared__ float b1s[HH], b2s[HH], b3s[FF];

  const int tid = threadIdx.x;
  const int R0  = blockIdx.x * 32;

  for (int i = tid; i < 32 * HH; i += 256) aggT[i] = agg[R0 * HH + i];

  const int lane = tid & 31, wave = tid >> 5;
  const int half = lane >> 4, l15 = lane & 15;
  const int m0 = (wave >> 2) * 16;   // 2 M-tiles
  const int n0 = (wave & 3) * 16;    // 4 N-tiles

  for (int path = 0; path < 2; ++path) {
    const float* W1p = path ? vW1 : mW1;
    const float* W2p = path ? vW2 : mW2;
    const float* W3p = path ? vW3 : mW3;
    const float* B1p = path ? vB1 : mB1;
    const float* B2p = path ? vB2 : mB2;
    const float* B3p = path ? vB3 : mB3;

    __syncthreads();  // aggT ready / previous path done with Wa,Wb
    for (int i = tid; i < HH * HH; i += 256) { Wa[i] = W1p[i]; Wb[i] = W2p[i]; }
    for (int i = tid; i < HH * FF; i += 256) W3s[i] = W3p[i];
    if (tid < HH) { b1s[tid] = B1p[tid]; b2s[tid] = B2p[tid]; }
    if (tid < FF) b3s[tid] = B3p[tid];
    __syncthreads();

    // layer 1: aggT @ Wa (+b1, relu) -> hA
    {
      v8f acc = {};
#pragma unroll
      for (int kk = 0; kk < 16; ++kk) {
        int k0 = kk * 4 + half * 2;
        int row = m0 + l15;
        v2f a; a.x = aggT[row * HH + k0]; a.y = aggT[row * HH + k0 + 1];
        v2f b; b.x = Wa[k0 * HH + n0 + l15]; b.y = Wa[(k0 + 1) * HH + n0 + l15];
        acc = wmma4(a, b, acc);
      }
#pragma unroll
      for (int r = 0; r < 8; ++r) {
        int row = m0 + r + half * 8;
        int col = n0 + l15;
        hA[row * HH + col] = fmaxf(acc[r] + b1s[col], 0.f);
      }
    }
    __syncthreads();

    // layer 2: hA @ Wb (+b2, relu) -> hB
    {
      v8f acc = {};
#pragma unroll
      for (int kk = 0; kk < 16; ++kk) {
        int k0 = kk * 4 + half * 2;
        int row = m0 + l15;
        v2f a; a.x = hA[row * HH + k0]; a.y = hA[row * HH + k0 + 1];
        v2f b; b.x = Wb[k0 * HH + n0 + l15]; b.y = Wb[(k0 + 1) * HH + n0 + l15];
        acc = wmma4(a, b, acc);
      }
#pragma unroll
      for (int r = 0; r < 8; ++r) {
        int row = m0 + r + half * 8;
        int col = n0 + l15;
        hB[row * HH + col] = fmaxf(acc[r] + b2s[col], 0.f);
      }
    }
    __syncthreads();

    // head: 64 -> 4, one output per thread (128 threads)
    if (tid < 32 * FF) {
      int row = tid >> 2, f = tid & 3;
      float s = b3s[f];
#pragma unroll 4
      for (int k = 0; k < HH; ++k) s += hB[row * HH + k] * W3s[k * FF + f];

      int rflat = R0 + row;
      int bc = rflat / NN, n = rflat % NN;
      int bs = bc >> 6, c = bc & 63;
      int t = 2 * c + step;

      if (path == 0) {
        prevNext[(bc * NN + n) * FF + f] = s;  // feeds step 1
        if (t < 127) out[((bs * 127 + t) * NN + n) * FF + f] = s;
      } else {
        float sp = (s > 20.f) ? s : log1pf(expf(s));
        sp = fminf(fmaxf(sp, 1e-8f), 100.f);
        if (t < 127) out[OUT_HALF + ((bs * 127 + t) * NN + n) * FF + f] = sp;
      }
    }
    __syncthreads();
  }
}

// ---------------------------------------------------------------------------
extern "C" void kernel_launch(void* const* d_in, const int* in_sizes, int n_in,
                              void* d_out, int out_size, void* d_ws, size_t ws_size,
                              hipStream_t stream) {
  const float* data  = (const float*)d_in[0];
  const float* graph = (const float*)d_in[1];
  const float* msgW1 = (const float*)d_in[2];
  const float* msgB1 = (const float*)d_in[3];
  const float* msgW2 = (const float*)d_in[4];
  const float* msgB2 = (const float*)d_in[5];
  const float* mW1 = (const float*)d_in[6];
  const float* mB1 = (const float*)d_in[7];
  const float* mW2 = (const float*)d_in[8];
  const float* mB2 = (const float*)d_in[9];
  const float* mW3 = (const float*)d_in[10];
  const float* mB3 = (const float*)d_in[11];
  const float* vW1 = (const float*)d_in[12];
  const float* vB1 = (const float*)d_in[13];
  const float* vW2 = (const float*)d_in[14];
  const float* vB2 = (const float*)d_in[15];
  const float* vW3 = (const float*)d_in[16];
  const float* vB3 = (const float*)d_in[17];
  float* out = (float*)d_out;

  float* prevA = (float*)d_ws;                 // 2048*30*4
  float* prevB = prevA + BCC * NN * FF;        // 2048*30*4
  float* aggG  = prevB + BCC * NN * FF;        // 2048*30*64 (15.7 MB, L2-resident)

  prep_prev<<<(BCC * NN * FF + 255) / 256, 256, 0, stream>>>(data, prevA);

  // step 0
  edge_agg<<<BCC, 256, 0, stream>>>(prevA, graph, msgW1, msgB1, msgW2, msgB2, aggG);
  node_mlp<<<ROWS / 32, 256, 0, stream>>>(aggG, mW1, mB1, mW2, mB2, mW3, mB3,
                                          vW1, vB1, vW2, vB2, vW3, vB3,
                                          out, prevB, 0);
  // step 1 (prev = mu of step 0)
  edge_agg<<<BCC, 256, 0, stream>>>(prevB, graph, msgW1, msgB1, msgW2, msgB2, aggG);
  node_mlp<<<ROWS / 32, 256, 0, stream>>>(aggG, mW1, mB1, mW2, mB2, mW3, mB3,
                                          vW1, vB1, vW2, vB2, vW3, vB3,
                                          out, prevA, 1);
}